// ContMix_23673859735870
// MI455X (gfx1250) — compile-verified
//
#include <hip/hip_runtime.h>
#include <hip/hip_bf16.h>
#include <math.h>

#define CDIM 64
#define HH 64
#define WW 64
#define EPSB 1e-5f

typedef float v2f __attribute__((ext_vector_type(2)));
typedef float v8f __attribute__((ext_vector_type(8)));

// D = A(16x4,f32) * B(4x16,f32) + C(16x16,f32)
__device__ __forceinline__ v8f wmma4(v2f a, v2f b, v8f c) {
  return __builtin_amdgcn_wmma_f32_16x16x4_f32(false, a, false, b, (short)0, c,
                                               false, false);
}

// ---------------------------------------------------------------------------
// Kernel 1: pooled k + bn + fold with proj_w  ->  pk[b][g][o(pad80)][cc]
// ---------------------------------------------------------------------------
__global__ __launch_bounds__(256) void kern_pk(
    const float* __restrict__ x, const float* __restrict__ wk_w,
    const float* __restrict__ wk_g, const float* __restrict__ wk_b,
    const float* __restrict__ wk_m, const float* __restrict__ wk_v,
    const float* __restrict__ proj_w, float* __restrict__ pk) {
  __shared__ float pooled[32 * 49];
  __shared__ float kmat[32 * 49];
  __shared__ float lwk[32 * 32];
  __shared__ float lproj[74 * 49];
  int b = blockIdx.x, t = threadIdx.x;
  for (int i = t; i < 32 * 32; i += 256) lwk[i] = wk_w[i];
  for (int i = t; i < 74 * 49; i += 256) lproj[i] = proj_w[i];
  // separable 7x7 average pool of kctx channels (32..63)
  for (int item = t; item < 32 * 49; item += 256) {
    int ci = item / 49, pq = item % 49, pp = pq / 7, qq = pq % 7;
    int sp = pp * HH / 7, ep = ((pp + 1) * HH + 6) / 7;
    int sq = qq * WW / 7, eq = ((qq + 1) * WW + 6) / 7;
    const float* xp = x + ((size_t)b * CDIM + 32 + ci) * HH * WW;
    float s = 0.f;
    for (int hh = sp; hh < ep; ++hh)
      for (int ww = sq; ww < eq; ++ww) s += xp[hh * WW + ww];
    pooled[item] = s / (float)((ep - sp) * (eq - sq));
  }
  __syncthreads();
  // k = bn(wk . pooled)
  for (int item = t; item < 32 * 49; item += 256) {
    int co = item / 49, pq = item % 49;
    float s = 0.f;
    for (int ci = 0; ci < 32; ++ci) s += lwk[co * 32 + ci] * pooled[ci * 49 + pq];
    float sc = wk_g[co] * rsqrtf(wk_v[co] + EPSB);
    kmat[item] = (s - wk_m[co]) * sc + wk_b[co];
  }
  __syncthreads();
  // pk[g][o][cc] = sum_l proj_w[o][l] * k[g*8+cc][l]  (o padded to 80, zeros)
  for (int item = t; item < 4 * 80 * 8; item += 256) {
    int g = item / 640, rem = item % 640, o = rem / 8, cc = rem % 8;
    float s = 0.f;
    if (o < 74) {
      const float* pr = &lproj[o * 49];
      const float* kr = &kmat[(g * 8 + cc) * 49];
      for (int l = 0; l < 49; ++l) s += pr[l] * kr[l];
    }
    pk[((size_t)(b * 4 + g) * 80 + o) * 8 + cc] = s;
  }
}

// ---------------------------------------------------------------------------
// Kernel 2: q-conv (WMMA) -> per-head logits (WMMA) -> rpb + softmax ->
//           neighborhood attention aggregation. Tile = 8x16 pixels (128 px).
// ---------------------------------------------------------------------------
__global__ __launch_bounds__(256) void kern_attn(
    const float* __restrict__ x, const float* __restrict__ wq_w,
    const float* __restrict__ wq_g, const float* __restrict__ wq_b,
    const float* __restrict__ wq_m, const float* __restrict__ wq_v,
    const float* __restrict__ proj_b, const float* __restrict__ rpb1,
    const float* __restrict__ rpb2, const float* __restrict__ pk,
    float* __restrict__ mid) {
  __shared__ float lq[32 * 128];   // q (channel-major)
  __shared__ float lu[128 * 74];   // phase1: xq (32x128); later: logits rows
  __shared__ float lwq[32 * 32];
  __shared__ float lr1[4 * 81];
  __shared__ float lr2[4 * 169];
  int b = blockIdx.y, tile = blockIdx.x, t = threadIdx.x;
  int h0 = (tile >> 2) * 8, w0 = (tile & 3) * 16;
  int lane = t & 31, wv = t >> 5;
  int l15 = lane & 15, hi = lane >> 4;

  for (int i = t; i < 32 * 32; i += 256) lwq[i] = wq_w[i];
  for (int i = t; i < 4 * 81; i += 256) lr1[i] = rpb1[i];
  for (int i = t; i < 4 * 169; i += 256) lr2[i] = rpb2[i];
  for (int i = t; i < 32 * 128; i += 256) {
    int ci = i >> 7, p = i & 127;
    lu[ci * 128 + p] =
        x[(((size_t)b * CDIM + ci) * HH + h0 + (p >> 4)) * WW + w0 + (p & 15)];
  }
  __syncthreads();

  // q = bn(wq . x) * 0.25 ; GEMM M=128 N=32 K=32
  for (int it = 0; it < 2; ++it) {
    int id = wv * 2 + it;
    int m0 = (id & 7) * 16, n0 = (id >> 3) * 16;
    v8f acc = {0, 0, 0, 0, 0, 0, 0, 0};
    for (int k0 = 0; k0 < 32; k0 += 4) {
      int ks = k0 + hi * 2;
      v2f a, bf;
      a.x = lu[ks * 128 + m0 + l15];
      a.y = lu[(ks + 1) * 128 + m0 + l15];
      bf.x = lwq[(n0 + l15) * 32 + ks];
      bf.y = lwq[(n0 + l15) * 32 + ks + 1];
      acc = wmma4(a, bf, acc);
    }
    int co = n0 + l15;
    float sc = wq_g[co] * rsqrtf(wq_v[co] + EPSB);
    float mn = wq_m[co], bb = wq_b[co];
    for (int rr = 0; rr < 8; ++rr) {
      int m = m0 + rr + hi * 8;
      lq[co * 128 + m] = ((acc[rr] - mn) * sc + bb) * 0.25f;
    }
  }
  __syncthreads();

  for (int g = 0; g < 4; ++g) {
    const float* pkg = pk + (size_t)(b * 4 + g) * 640;
    // logits = q_g . pk_g + proj_b ; GEMM M=128 N=80(74) K=8
    for (int it = 0; it < 5; ++it) {
      int id = wv * 5 + it;
      int m0 = (id % 8) * 16, n0 = (id / 8) * 16;
      v8f acc = {0, 0, 0, 0, 0, 0, 0, 0};
      for (int k0 = 0; k0 < 8; k0 += 4) {
        int ks = k0 + hi * 2;
        v2f a, bf;
        a.x = lq[(g * 8 + ks) * 128 + m0 + l15];
        a.y = lq[(g * 8 + ks + 1) * 128 + m0 + l15];
        bf.x = pkg[(n0 + l15) * 8 + ks];
        bf.y = pkg[(n0 + l15) * 8 + ks + 1];
        acc = wmma4(a, bf, acc);
      }
      int o = n0 + l15;
      if (o < 74) {
        float pb = proj_b[o];
        for (int rr = 0; rr < 8; ++rr) {
          int m = m0 + rr + hi * 8;
          lu[m * 74 + o] = acc[rr] + pb;
        }
      }
    }
    __syncthreads();

    // waves 0-3: 5x5 branch for pixel t; waves 4-7: 7x7 branch for pixel t-128
    {
      int p = t & 127;
      int r = p >> 4, cpx = p & 15;
      int h = h0 + r, w = w0 + cpx;
      float* row = &lu[p * 74];
      if (t < 128) {
        int ihh = (h < 2) ? h : ((h >= 62) ? h - 59 : 2);
        int iww = (w < 2) ? w : ((w >= 62) ? w - 59 : 2);
        int ro = 4 - ihh, co_ = 4 - iww;
        const float* rp = &lr1[g * 81];
        float mx = -1e30f;
        for (int i = 0; i < 5; ++i)
          for (int j = 0; j < 5; ++j)
            mx = fmaxf(mx, row[i * 5 + j] + rp[(ro + i) * 9 + co_ + j]);
        float s = 0.f;
        for (int i = 0; i < 5; ++i)
          for (int j = 0; j < 5; ++j) {
            float e = __expf(row[i * 5 + j] + rp[(ro + i) * 9 + co_ + j] - mx);
            row[i * 5 + j] = e;
            s += e;
          }
        float inv = 1.f / s;
        int si = min(max(h - 2, 0), 59), sj = min(max(w - 2, 0), 59);
        const float* xb = x + ((size_t)b * CDIM + g * 8) * HH * WW;
        float acc[8] = {0, 0, 0, 0, 0, 0, 0, 0};
        for (int i = 0; i < 5; ++i)
          for (int j = 0; j < 5; ++j) {
            float wgt = row[i * 5 + j] * inv;
            const float* px = xb + (si + i) * WW + sj + j;
            for (int cc = 0; cc < 8; ++cc) acc[cc] += wgt * px[cc * HH * WW];
          }
        for (int cc = 0; cc < 8; ++cc)
          mid[(((size_t)b * CDIM + g * 8 + cc) * HH + h) * WW + w] = acc[cc];
      } else {
        int ihh = (h < 3) ? h : ((h >= 61) ? h - 57 : 3);
        int iww = (w < 3) ? w : ((w >= 61) ? w - 57 : 3);
        int ro = 6 - ihh, co_ = 6 - iww;
        const float* rp = &lr2[g * 169];
        float mx = -1e30f;
        for (int i = 0; i < 7; ++i)
          for (int j = 0; j < 7; ++j)
            mx = fmaxf(mx, row[25 + i * 7 + j] + rp[(ro + i) * 13 + co_ + j]);
        float s = 0.f;
        for (int i = 0; i < 7; ++i)
          for (int j = 0; j < 7; ++j) {
            float e =
                __expf(row[25 + i * 7 + j] + rp[(ro + i) * 13 + co_ + j] - mx);
            row[25 + i * 7 + j] = e;
            s += e;
          }
        float inv = 1.f / s;
        int si = min(max(h - 3, 0), 57), sj = min(max(w - 3, 0), 57);
        const float* xb = x + ((size_t)b * CDIM + 32 + g * 8) * HH * WW;
        float acc[8] = {0, 0, 0, 0, 0, 0, 0, 0};
        for (int i = 0; i < 7; ++i)
          for (int j = 0; j < 7; ++j) {
            float wgt = row[25 + i * 7 + j] * inv;
            const float* px = xb + (si + i) * WW + sj + j;
            for (int cc = 0; cc < 8; ++cc) acc[cc] += wgt * px[cc * HH * WW];
          }
        for (int cc = 0; cc < 8; ++cc)
          mid[(((size_t)b * CDIM + 32 + g * 8 + cc) * HH + h) * WW + w] = acc[cc];
      }
    }
    __syncthreads();
  }
}

// ---------------------------------------------------------------------------
// Kernel 3: y = bn(dy_w . mid) ; per-tile GEMM M=128 N=64 K=64 via WMMA
// ---------------------------------------------------------------------------
__global__ __launch_bounds__(256) void kern_out(
    const float* __restrict__ mid, const float* __restrict__ dy_w,
    const float* __restrict__ dy_g, const float* __restrict__ dy_b,
    const float* __restrict__ dy_m, const float* __restrict__ dy_v,
    float* __restrict__ out) {
  __shared__ float la[64 * 128];
  __shared__ float lw[64 * 64];
  int b = blockIdx.y, tile = blockIdx.x, t = threadIdx.x;
  int h0 = (tile >> 2) * 8, w0 = (tile & 3) * 16;
  int lane = t & 31, wv = t >> 5;
  int l15 = lane & 15, hi = lane >> 4;
  for (int i = t; i < 64 * 64; i += 256) lw[i] = dy_w[i];
  for (int i = t; i < 64 * 128; i += 256) {
    int ci = i >> 7, p = i & 127;
    la[ci * 128 + p] =
        mid[(((size_t)b * CDIM + ci) * HH + h0 + (p >> 4)) * WW + w0 + (p & 15)];
  }
  __syncthreads();
  for (int it = 0; it < 4; ++it) {
    int id = wv * 4 + it;
    int m0 = (id & 7) * 16, n0 = (id >> 3) * 16;
    v8f acc = {0, 0, 0, 0, 0, 0, 0, 0};
    for (int k0 = 0; k0 < 64; k0 += 4) {
      int ks = k0 + hi * 2;
      v2f a, bf;
      a.x = la[ks * 128 + m0 + l15];
      a.y = la[(ks + 1) * 128 + m0 + l15];
      bf.x = lw[(n0 + l15) * 64 + ks];
      bf.y = lw[(n0 + l15) * 64 + ks + 1];
      acc = wmma4(a, bf, acc);
    }
    int co = n0 + l15;
    float sc = dy_g[co] * rsqrtf(dy_v[co] + EPSB);
    float mn = dy_m[co], bb = dy_b[co];
    for (int rr = 0; rr < 8; ++rr) {
      int m = m0 + rr + hi * 8;
      out[(((size_t)b * CDIM + co) * HH + h0 + (m >> 4)) * WW + w0 + (m & 15)] =
          (acc[rr] - mn) * sc + bb;
    }
  }
}

extern "C" void kernel_launch(void* const* d_in, const int* in_sizes, int n_in,
                              void* d_out, int out_size, void* d_ws,
                              size_t ws_size, hipStream_t stream) {
  const float* x = (const float*)d_in[0];
  const float* wq_w = (const float*)d_in[1];
  const float* wq_g = (const float*)d_in[2];
  const float* wq_b = (const float*)d_in[3];
  const float* wq_m = (const float*)d_in[4];
  const float* wq_v = (const float*)d_in[5];
  const float* wk_w = (const float*)d_in[6];
  const float* wk_g = (const float*)d_in[7];
  const float* wk_b = (const float*)d_in[8];
  const float* wk_m = (const float*)d_in[9];
  const float* wk_v = (const float*)d_in[10];
  const float* proj_w = (const float*)d_in[11];
  const float* proj_b = (const float*)d_in[12];
  const float* rpb1 = (const float*)d_in[13];
  const float* rpb2 = (const float*)d_in[14];
  const float* dy_w = (const float*)d_in[15];
  const float* dy_g = (const float*)d_in[16];
  const float* dy_b = (const float*)d_in[17];
  const float* dy_m = (const float*)d_in[18];
  const float* dy_v = (const float*)d_in[19];

  float* pk = (float*)d_ws;                 // 8*4*80*8 floats
  float* mid = pk + (size_t)8 * 4 * 80 * 8; // 8*64*64*64 floats

  kern_pk<<<8, 256, 0, stream>>>(x, wk_w, wk_g, wk_b, wk_m, wk_v, proj_w, pk);
  dim3 grid2(32, 8);
  kern_attn<<<grid2, 256, 0, stream>>>(x, wq_w, wq_g, wq_b, wq_m, wq_v, proj_b,
                                       rpb1, rpb2, pk, mid);
  kern_out<<<grid2, 256, 0, stream>>>(mid, dy_w, dy_g, dy_b, dy_m, dy_v,
                                      (float*)d_out);
}